// TemporalAttention_54322746359850
// MI455X (gfx1250) — compile-verified
//
#include <hip/hip_runtime.h>

// Fused "temporal attention" block for MI455X (gfx1250).
// Key identity: softmax over the last axis sums to 1, and the einsum
// 'TNS,BNH->BNH' reduces attn over both T and S  =>  y = N * v.
// So the whole graph collapses to a chain of 1024x{128,256} x {128} GEMMs,
// run as per-wave 16x16 f32 WMMA tiles, fully fused in one kernel.
// Stage-0 stripe loads use GLOBAL_LOAD_ASYNC_TO_LDS_B128 (ASYNCcnt path),
// overlapped with global_prefetch warm-up of the weight matrices in L2.

namespace {

constexpr int NN  = 1024;   // nodes
constexpr int H   = 128;    // hidden
constexpr int LDA = 132;    // padded LDS row stride (floats) -> no bank conflicts

typedef __attribute__((ext_vector_type(2))) float v2f;
typedef __attribute__((ext_vector_type(8))) float v8f;

// One 16x16 output tile:  acc += A(16xH, LDS, stride LDA) * B(Hx128, global)[:, n0:n0+16]
// WMMA f32 16x16x4 layouts (ISA 7.12.2):
//   A: lane m = lane&15, K pair = 2*(lane>>4) + {0,1}   (contiguous -> b64 LDS load)
//   B: lane n = lane&15, same K split                   (coalesced b32 global loads)
//   C/D: VGPR r holds (row = r + 8*(lane>>4), col = lane&15)
__device__ __forceinline__ v8f tile_gemm(const float* __restrict__ As,
                                         const float* __restrict__ Bg,
                                         int n0, int lane, v8f acc) {
  const int m  = lane & 15;
  const int hi = lane >> 4;
  const int n  = lane & 15;
  const float* arow = As + m * LDA + 2 * hi;          // 8B aligned
  const float* bcol = Bg + (2 * hi) * H + n0 + n;
#pragma unroll 4
  for (int k0 = 0; k0 < H; k0 += 4) {
    v2f a = *(const v2f*)(arow + k0);                 // ds_load_2addr_b64
    v2f b;
    b.x = bcol[(k0 + 0) * H];                         // global b32, 64B/half-wave
    b.y = bcol[(k0 + 1) * H];
    acc = __builtin_amdgcn_wmma_f32_16x16x4_f32(
        /*neg_a=*/false, a, /*neg_b=*/false, b,
        /*c_mod=*/(short)0, acc, /*reuse_a=*/false, /*reuse_b=*/false);
  }
  return acc;
}

__device__ __forceinline__ float sigmoidf_(float x) {
  return 1.0f / (1.0f + __expf(-x));
}

// Async copy of one 16B chunk: global -> LDS, tracked by ASYNCcnt.
// gfx1250 generic pointers to LDS carry the LDS byte offset in their low
// 32 bits (ISA 10.2 aperture mapping), which is exactly the VDST operand.
__device__ __forceinline__ void async_g2l_b128(void* lds_ptr, const void* gptr) {
  unsigned ldsoff = (unsigned)(size_t)lds_ptr;
  asm volatile("global_load_async_to_lds_b128 %0, %1, off"
               :: "v"(ldsoff), "v"(gptr) : "memory");
}

__global__ __launch_bounds__(256) void
temporal_attn_fused(const float* __restrict__ h_prev, const float* __restrict__ h_pred,
                    const float* __restrict__ Wv,  const float* __restrict__ bv,
                    const float* __restrict__ Wo,  const float* __restrict__ bo,
                    const float* __restrict__ gW1, const float* __restrict__ gb1,
                    const float* __restrict__ gW2, const float* __restrict__ gb2,
                    const float* __restrict__ mW1, const float* __restrict__ mb1,
                    const float* __restrict__ mW2, const float* __restrict__ mb2,
                    float* __restrict__ out) {
  __shared__ float sPred[16 * LDA];
  __shared__ float sPrev[16 * LDA];
  __shared__ float sY   [16 * LDA];
  __shared__ float sG1  [16 * LDA];
  __shared__ float sHc  [16 * LDA];
  __shared__ float sM1  [16 * LDA];

  const int r0   = blockIdx.x * 16;      // 16-row stripe of the 1024 nodes
  const int tid  = threadIdx.x;
  const int lane = tid & 31;
  const int wave = tid >> 5;             // 8 waves -> 8 column tiles of 16
  const int n0   = wave * 16;
  const int hi   = lane >> 4;
  const int n    = lane & 15;

  // ---- stage 0: async stripe DMA into LDS (ASYNCcnt), overlap L2 warm-up ----
  {
    const float* gp = h_pred + (size_t)r0 * H;
    const float* gv = h_prev + (size_t)r0 * H;
#pragma unroll
    for (int jj = 0; jj < 2; ++jj) {
      const int i   = tid + jj * 256;    // float4 index; 512 per 16x128 stripe
      const int row = i >> 5;            // 32 float4 per row
      const int col = (i & 31) * 4;
      async_g2l_b128(sPred + row * LDA + col, gp + (size_t)i * 4);
      async_g2l_b128(sPrev + row * LDA + col, gv + (size_t)i * 4);
    }
    // Warm the first weight matrices in L2 while the async DMA is in flight.
    __builtin_prefetch(Wv  + (size_t)tid * 64, 0, 3);
    __builtin_prefetch(gW1 + (size_t)tid * 64, 0, 3);
    __builtin_prefetch(gW1 + (size_t)H * H + (size_t)tid * 64, 0, 3);
    asm volatile("s_wait_asynccnt 0x0" ::: "memory");
  }
  __syncthreads();

  const v8f zero = {};

  // ---- stage 1: Y = 1024 * (h_pred @ Wv + bv)   (y = N * v collapse) ----
  v8f acc = tile_gemm(sPred, Wv, n0, lane, zero);
  {
    const float bvn = bv[n0 + n];
#pragma unroll
    for (int r = 0; r < 8; ++r)
      sY[(r + 8 * hi) * LDA + n0 + n] = 1024.0f * (acc[r] + bvn);
  }
  __syncthreads();

  // ---- stage 2: OUT = Y @ Wo + bo  (kept in registers) ----
  v8f accOut = tile_gemm(sY, Wo, n0, lane, zero);
  {
    const float bon = bo[n0 + n];
#pragma unroll
    for (int r = 0; r < 8; ++r) accOut[r] += bon;
  }

  // ---- stage 3: G1 = silu([h_pred,h_prev] @ gW1 + gb1) ----
  acc = tile_gemm(sPred, gW1,         n0, lane, zero);
  acc = tile_gemm(sPrev, gW1 + H * H, n0, lane, acc);
  {
    const float gb1n = gb1[n0 + n];
#pragma unroll
    for (int r = 0; r < 8; ++r) {
      const float x = acc[r] + gb1n;
      sG1[(r + 8 * hi) * LDA + n0 + n] = x * sigmoidf_(x);   // silu
    }
  }
  __syncthreads();

  // ---- stage 4: gate = sigmoid(G1 @ gW2 + gb2); h_corr = h_prev + gate*OUT ----
  acc = tile_gemm(sG1, gW2, n0, lane, zero);
  {
    const float gb2n = gb2[n0 + n];
#pragma unroll
    for (int r = 0; r < 8; ++r) {
      const int row = r + 8 * hi;
      const float g  = sigmoidf_(acc[r] + gb2n);
      const float hp = sPrev[row * LDA + n0 + n];
      sHc[row * LDA + n0 + n] = hp + g * accOut[r];
    }
  }
  __syncthreads();

  // ---- stage 5: M1 = relu([h_corr,h_prev] @ mW1 + mb1) ----
  acc = tile_gemm(sHc,  mW1,         n0, lane, zero);
  acc = tile_gemm(sPrev, mW1 + H * H, n0, lane, acc);
  {
    const float mb1n = mb1[n0 + n];
#pragma unroll
    for (int r = 0; r < 8; ++r) {
      const float x = acc[r] + mb1n;
      sM1[(r + 8 * hi) * LDA + n0 + n] = x > 0.0f ? x : 0.0f;
    }
  }
  __syncthreads();

  // ---- stage 6: mixed = h_corr + M1 @ mW2 + mb2  -> global ----
  acc = tile_gemm(sM1, mW2, n0, lane, zero);
  {
    const float mb2n = mb2[n0 + n];
#pragma unroll
    for (int r = 0; r < 8; ++r) {
      const int row = r + 8 * hi;
      const float v = sHc[row * LDA + n0 + n] + acc[r] + mb2n;
      out[(size_t)(r0 + row) * H + n0 + n] = v;
    }
  }
}

} // namespace

extern "C" void kernel_launch(void* const* d_in, const int* in_sizes, int n_in,
                              void* d_out, int out_size, void* d_ws, size_t ws_size,
                              hipStream_t stream) {
  // setup_inputs order:
  // 0 h_prev, 1 h_pred, 2 adj_rows, 3 adj_cols, 4 Wq, 5 bq, 6 Wk, 7 bk,
  // 8 Wv, 9 bv, 10 Wo, 11 bo, 12 gW1, 13 gb1, 14 gW2, 15 gb2,
  // 16 mW1, 17 mb1, 18 mW2, 19 mb2
  const float* h_prev = (const float*)d_in[0];
  const float* h_pred = (const float*)d_in[1];
  // adj/Wq/Wk are mathematically irrelevant: softmax rows sum to 1 and the
  // 'TNS,BNH->BNH' einsum reduces attn over T and S, so y == 1024 * v.
  const float* Wv  = (const float*)d_in[8];
  const float* bv  = (const float*)d_in[9];
  const float* Wo  = (const float*)d_in[10];
  const float* bo  = (const float*)d_in[11];
  const float* gW1 = (const float*)d_in[12];
  const float* gb1 = (const float*)d_in[13];
  const float* gW2 = (const float*)d_in[14];
  const float* gb2 = (const float*)d_in[15];
  const float* mW1 = (const float*)d_in[16];
  const float* mb1 = (const float*)d_in[17];
  const float* mW2 = (const float*)d_in[18];
  const float* mb2 = (const float*)d_in[19];
  float* out = (float*)d_out;

  dim3 grid(NN / 16);   // 64 stripes
  dim3 block(256);      // 8 wave32 -> 8 column tiles
  temporal_attn_fused<<<grid, block, 0, stream>>>(
      h_prev, h_pred, Wv, bv, Wo, bo, gW1, gb1, gW2, gb2, mW1, mb1, mW2, mb2, out);
}